// SimpleGCNLayer_67568425501458
// MI455X (gfx1250) — compile-verified
//
#include <hip/hip_runtime.h>

typedef __attribute__((ext_vector_type(2))) float v2f;
typedef __attribute__((ext_vector_type(8))) float v8f;

#define GCN_N_NODES 100000
#define GCN_N_EDGES 1600000
#define GCN_D 32

// ---------------------------------------------------------------------------
// Kernel 1: y = x @ W^T  via V_WMMA_F32_16X16X4_F32 (wave32, 16x16 tiles).
// One wave handles a 16-row tile of x and produces 16x32 of y:
//   2 output col-tiles (o=0..15, o=16..31) x 8 k-chunks (K=32, k=4 per WMMA).
//
// A (16x4 f32) lane layout (ISA 7.12.2): lane l: m = l&15, kb = (l>>4)*2,
//   VGPR0 = A[m][kb], VGPR1 = A[m][kb+1].
// B (4x16 f32): mirrored striping: lane l: n = l&15, kb = (l>>4)*2,
//   VGPR0 = B[kb][n], VGPR1 = B[kb+1][n];  B[k][n] = W[o0+n][kc*4+k].
// C/D (16x16 f32): VGPR r -> row r + (l>=16 ? 8 : 0), col l&15.
// ---------------------------------------------------------------------------
__global__ __launch_bounds__(256) void gcn_gemm_wmma(
    const float* __restrict__ x, const float* __restrict__ W,
    float* __restrict__ y, int n_tiles)
{
    const int lane = threadIdx.x & 31;
    const int wave = threadIdx.x >> 5;
    const int tile = blockIdx.x * 8 + wave;
    if (tile >= n_tiles) return;          // wave-uniform: EXEC stays all-ones

    const int m  = lane & 15;             // A row / B col / D col within tile
    const int kb = (lane >> 4) * 2;       // K sub-offset per half-wave

    // Preload B for both output column tiles (W is [32 out][32 in], row-major)
    v2f b0[8], b1[8];
#pragma unroll
    for (int kc = 0; kc < 8; ++kc) {
        const int k = kc * 4 + kb;
        const float* w0 = W + (m)      * GCN_D + k;   // out channels 0..15
        const float* w1 = W + (16 + m) * GCN_D + k;   // out channels 16..31
        v2f t0; t0.x = w0[0]; t0.y = w0[1];
        v2f t1; t1.x = w1[0]; t1.y = w1[1];
        b0[kc] = t0; b1[kc] = t1;
    }

    v8f c0 = {}; v8f c1 = {};
    const float* xr = x + (size_t)tile * 16 * GCN_D;
#pragma unroll
    for (int kc = 0; kc < 8; ++kc) {
        const int k = kc * 4 + kb;
        v2f a = *(const v2f*)(xr + m * GCN_D + k);   // 8B-aligned global_load_b64
        c0 = __builtin_amdgcn_wmma_f32_16x16x4_f32(
                 false, a, false, b0[kc], (short)0, c0, false, false);
        c1 = __builtin_amdgcn_wmma_f32_16x16x4_f32(
                 false, a, false, b1[kc], (short)0, c1, false, false);
    }

    // Store D tiles
    const int n     = lane & 15;
    const int rbase = (lane >> 4) * 8;
    float* yr = y + (size_t)tile * 16 * GCN_D;
#pragma unroll
    for (int r = 0; r < 8; ++r) {
        yr[(rbase + r) * GCN_D + n]      = c0[r];
        yr[(rbase + r) * GCN_D + 16 + n] = c1[r];
    }
}

// ---------------------------------------------------------------------------
// Kernel 2: zero the output (harness poisons d_out with 0xAA before timing).
// ---------------------------------------------------------------------------
__global__ __launch_bounds__(256) void gcn_zero(float4* __restrict__ out, int n4)
{
    int i = blockIdx.x * 256 + threadIdx.x;
    if (i < n4) {
        float4 z; z.x = 0.f; z.y = 0.f; z.z = 0.f; z.w = 0.f;
        out[i] = z;
    }
}

// ---------------------------------------------------------------------------
// Kernel 3: out[dst] += y[src], one wave per edge, lane = feature.
// y (12.8 MB) and out (12.8 MB) both live in the 192 MB L2, so the random
// gather + 51.2M global_atomic_add_f32 run at L2 speed, not HBM.
// edge_index layout: [2, E] flat -> src = ei[e], dst = ei[E + e].
// ---------------------------------------------------------------------------
__global__ __launch_bounds__(256) void gcn_scatter(
    const float* __restrict__ y, const int* __restrict__ ei,
    float* __restrict__ out, int n_edges)
{
    int t = blockIdx.x * 256 + threadIdx.x;
    int e = t >> 5;
    int f = t & 31;
    if (e >= n_edges) return;
    int src = ei[e];
    int dst = ei[n_edges + e];
    atomicAdd(out + (size_t)dst * GCN_D + f, y[(size_t)src * GCN_D + f]);
}

// ---------------------------------------------------------------------------
// Host launch: inputs in setup_inputs() order: x [N,32] f32, edge_index [2,E]
// int, W [32,32] f32. d_out: [N,32] f32. d_ws holds y = x @ W^T (12.8 MB).
// ---------------------------------------------------------------------------
extern "C" void kernel_launch(void* const* d_in, const int* in_sizes, int n_in,
                              void* d_out, int out_size, void* d_ws, size_t ws_size,
                              hipStream_t stream)
{
    const float* x  = (const float*)d_in[0];
    const int*   ei = (const int*)d_in[1];
    const float* W  = (const float*)d_in[2];
    float*       out = (float*)d_out;
    float*       y   = (float*)d_ws;       // N_NODES * 32 floats = 12.8 MB

    const int n_tiles = GCN_N_NODES / 16;  // 100000 / 16 = 6250, exact
    const int gemm_blocks = (n_tiles + 7) / 8;
    gcn_gemm_wmma<<<gemm_blocks, 256, 0, stream>>>(x, W, y, n_tiles);

    const int n4 = GCN_N_NODES * GCN_D / 4;          // 800000 float4
    gcn_zero<<<(n4 + 255) / 256, 256, 0, stream>>>((float4*)out, n4);

    const long long nthreads = (long long)GCN_N_EDGES * 32;   // 51.2M
    const int scat_blocks = (int)((nthreads + 255) / 256);
    gcn_scatter<<<scat_blocks, 256, 0, stream>>>(y, ei, out, GCN_N_EDGES);
}